// QHilbertParam_9457517985933
// MI455X (gfx1250) — compile-verified
//
#include <hip/hip_runtime.h>
#include <math.h>

// ---------------------------------------------------------------------------
// Quantum circuit: out[B,8] from x[B,24], theta[3,6,3].
// s_enc (product state, VALU) -> psi = W s_enc (f32 WMMA GEMM [Bx128]x[128x128])
// -> Pauli expectations (lane reductions).
// Each wave processes TWO 16-row batch tiles per B-fragment load to halve the
// LDS-read : WMMA ratio (B frags are batch-invariant).
// ---------------------------------------------------------------------------

typedef float v2f __attribute__((ext_vector_type(2)));
typedef float v8f __attribute__((ext_vector_type(8)));

#define PAD 132                       // LDS row stride (floats): 4-bank shift, 16B aligned
#define PI_F 3.14159265358979323846f

struct cplx { float x, y; };
__device__ __forceinline__ cplx cmul(cplx a, cplx b) {
    return { a.x * b.x - a.y * b.y, a.x * b.y + a.y * b.x };
}
__device__ __forceinline__ cplx cadd(cplx a, cplx b) { return { a.x + b.x, a.y + b.y }; }

// ---------------------------------------------------------------------------
// Kernel 1: build G[128][128] (real embedding of the 64x64 layer unitary W).
// One block, 64 threads; thread t evolves basis column |t> in LDS.
// y = G u with u = [Re s; Im s], y = [Re psi; Im psi];  G = [[Wr,-Wi],[Wi,Wr]].
// ---------------------------------------------------------------------------
__global__ __launch_bounds__(64) void build_G_kernel(const float* __restrict__ theta,
                                                     float* __restrict__ G) {
    __shared__ cplx col[64 * 64];     // [column][amp]
    const int t = threadIdx.x;        // 0..63
    cplx* c = &col[t * 64];
    for (int i = 0; i < 64; ++i) c[i] = { 0.f, 0.f };
    c[t] = { 1.f, 0.f };

    auto apply1q = [&](int q, cplx g00, cplx g01, cplx g10, cplx g11) {
        const int m = 1 << (5 - q);
        for (int j = 0; j < 64; ++j) {
            if (j & m) continue;
            cplx a = c[j], b = c[j | m];
            c[j]     = cadd(cmul(g00, a), cmul(g01, b));
            c[j | m] = cadd(cmul(g10, a), cmul(g11, b));
        }
    };
    auto cnot = [&](int cc, int tt) {
        const int mc = 1 << (5 - cc), mt = 1 << (5 - tt);
        for (int j = 0; j < 64; ++j) {
            if ((j & mc) && !(j & mt)) {
                cplx tmp = c[j]; c[j] = c[j | mt]; c[j | mt] = tmp;
            }
        }
    };

    for (int l = 0; l < 3; ++l) {
        for (int q = 0; q < 5; ++q) cnot(q, q + 1);
        cnot(5, 0);
        for (int q = 0; q < 6; ++q) {
            const float ax = theta[(l * 6 + q) * 3 + 0];
            const float ay = theta[(l * 6 + q) * 3 + 1];
            const float az = theta[(l * 6 + q) * 3 + 2];
            float cx = cosf(0.5f * ax), sx = sinf(0.5f * ax);
            apply1q(q, {cx, 0.f}, {0.f, -sx}, {0.f, -sx}, {cx, 0.f});       // RX
            float cy = cosf(0.5f * ay), sy = sinf(0.5f * ay);
            apply1q(q, {cy, 0.f}, {-sy, 0.f}, {sy, 0.f}, {cy, 0.f});        // RY
            float cz = cosf(0.5f * az), sz = sinf(0.5f * az);
            apply1q(q, {cz, -sz}, {0.f, 0.f}, {0.f, 0.f}, {cz, sz});        // RZ
        }
    }
    __syncthreads();

    // W[row][colm] = col[colm][row]
    for (int i = 0; i < 256; ++i) {
        const int idx = t + 64 * i;          // 16384 = 128*128
        const int n = idx >> 7, k = idx & 127;
        float v;
        if (n < 64) v = (k < 64) ? col[k * 64 + n].x : -col[(k - 64) * 64 + n].y;
        else        v = (k < 64) ? col[k * 64 + (n - 64)].y : col[(k - 64) * 64 + (n - 64)].x;
        G[idx] = v;
    }
}

// ---------------------------------------------------------------------------
// Kernel 2: fused encode -> WMMA GEMM -> measurements.
// 128 threads = 4 waves; each wave owns 32 batch elements (2 row tiles).
// ---------------------------------------------------------------------------
__global__ __launch_bounds__(128) void qsim_kernel(const float* __restrict__ x,
                                                   const float* __restrict__ G,
                                                   float* __restrict__ out) {
    __shared__ float Mlds[128 * PAD];        // G[n][k] padded        (~66 KB)
    __shared__ float Ulds[4 * 32 * PAD];     // per-wave A panels     (~66 KB)

    const int tid  = threadIdx.x;
    const int w    = tid >> 5;               // wave id 0..3
    const int lane = tid & 31;
    const int l15  = lane & 15;
    const int h    = lane >> 4;              // half-wave

    // ---- stage G into LDS (float4, conflict-free via PAD=132) ----
    const float4* G4 = (const float4*)G;
    #pragma unroll
    for (int i = 0; i < 32; ++i) {
        const int idx4 = tid + 128 * i;      // 4096 float4 = 128x128
        const int n = idx4 >> 5, k4 = idx4 & 31;
        *(float4*)&Mlds[n * PAD + k4 * 4] = G4[idx4];
    }

    // ---- angle encoding: one lane per element; u row = [Re s(64) | Im s(64)] ----
    const int e = blockIdx.x * 128 + w * 32 + lane;
    const float* xp = x + e * 24;
    float vr[6][2], vi[6][2];
    #pragma unroll
    for (int q = 0; q < 6; ++q) {
        float m = (xp[4 * q] + xp[4 * q + 1] + xp[4 * q + 2] + xp[4 * q + 3]) * 0.25f;
        m = fminf(6.f, fmaxf(-6.f, m));
        const float a  = m * (PI_F / 6.f);
        const float c2 = cosf(0.5f * a),  s2 = sinf(0.5f * a);
        const float c4 = cosf(0.25f * a), s4 = sinf(0.25f * a);
        vr[q][0] =  c2 * c4;  vi[q][0] = -c2 * s4;   // RZ(a/2)·RX(a)|0>
        vr[q][1] =  s2 * s4;  vi[q][1] = -s2 * c4;
    }
    float* u = &Ulds[(w * 32 + lane) * PAD];
    for (int j = 0; j < 64; ++j) {
        float pr = vr[0][(j >> 5) & 1], pim = vi[0][(j >> 5) & 1];
        #pragma unroll
        for (int q = 1; q < 6; ++q) {
            const int b = (j >> (5 - q)) & 1;
            const float qr = vr[q][b], qi = vi[q][b];
            const float nr = pr * qr - pim * qi;
            pim = pr * qi + pim * qr;
            pr  = nr;
        }
        u[j] = pr; u[64 + j] = pim;
    }
    __syncthreads();

    // ---- GEMM: two 16x128 row tiles share each B fragment ----
    v8f acc[2][8];
    const v8f z8 = {0.f, 0.f, 0.f, 0.f, 0.f, 0.f, 0.f, 0.f};
    #pragma unroll
    for (int rt = 0; rt < 2; ++rt)
        #pragma unroll
        for (int ct = 0; ct < 8; ++ct) acc[rt][ct] = z8;

    const float* urow0 = &Ulds[(w * 32 +  0 + l15) * PAD];
    const float* urow1 = &Ulds[(w * 32 + 16 + l15) * PAD];
    for (int kt = 0; kt < 32; ++kt) {
        const int kb = kt * 4 + 2 * h;       // A/B layout: v0=K{0,2}, v1=K{1,3}
        v2f b[8];
        #pragma unroll
        for (int ct = 0; ct < 8; ++ct) {
            const float* brow = &Mlds[(ct * 16 + l15) * PAD + kb];   // B[k][n]=G[n][k]
            b[ct].x = brow[0]; b[ct].y = brow[1];
        }
        v2f a0; a0.x = urow0[kb]; a0.y = urow0[kb + 1];
        v2f a1; a1.x = urow1[kb]; a1.y = urow1[kb + 1];
        #pragma unroll
        for (int ct = 0; ct < 8; ++ct) {
            acc[0][ct] = __builtin_amdgcn_wmma_f32_16x16x4_f32(
                false, a0, false, b[ct], (short)0, acc[0][ct], false, false);
            acc[1][ct] = __builtin_amdgcn_wmma_f32_16x16x4_f32(
                false, a1, false, b[ct], (short)0, acc[1][ct], false, false);
        }
    }

    // ---- measurements ----
    // D layout: vgpr r -> rows {r, r+8}; lane l15 in tile ct -> col 16*ct+l15.
    // Cols 0..63 = Re psi, 64..127 = Im psi; lane holds amps j in {l15, l15+16, l15+32, l15+48}.
    auto rsum16 = [](float v) {              // reduce across 16-lane half-wave
        v += __shfl_xor(v, 1, 32);
        v += __shfl_xor(v, 2, 32);
        v += __shfl_xor(v, 4, 32);
        v += __shfl_xor(v, 8, 32);
        return v;
    };

    #pragma unroll
    for (int rt = 0; rt < 2; ++rt) {
        #pragma unroll
        for (int r = 0; r < 8; ++r) {
            float re[4], im[4];
            #pragma unroll
            for (int t4 = 0; t4 < 4; ++t4) { re[t4] = acc[rt][t4][r]; im[t4] = acc[rt][4 + t4][r]; }

            float z0 = 0.f, z2 = 0.f, z4 = 0.f, x1 = 0.f, x3 = 0.f, x5 = 0.f;
            #pragma unroll
            for (int t4 = 0; t4 < 4; ++t4) {
                const int j = t4 * 16 + l15;
                const float p = re[t4] * re[t4] + im[t4] * im[t4];
                z0 += ((j >> 5) & 1) ? -p : p;          // Z qubit0 (mask 32)
                z2 += ((j >> 3) & 1) ? -p : p;          // Z qubit2 (mask 8)
                z4 += ((j >> 1) & 1) ? -p : p;          // Z qubit4 (mask 2)
                x1 += re[t4] * re[t4 ^ 1] + im[t4] * im[t4 ^ 1];          // X q1 (mask 16)
                x3 += re[t4] * __shfl_xor(re[t4], 4, 32)
                    + im[t4] * __shfl_xor(im[t4], 4, 32);                 // X q3 (mask 4)
                x5 += re[t4] * __shfl_xor(re[t4], 1, 32)
                    + im[t4] * __shfl_xor(im[t4], 1, 32);                 // X q5 (mask 1)
            }
            z0 = rsum16(z0); z2 = rsum16(z2); z4 = rsum16(z4);
            x1 = rsum16(x1); x3 = rsum16(x3); x5 = rsum16(x5);

            if (l15 == 0) {
                const int ee = blockIdx.x * 128 + w * 32 + rt * 16 + r + h * 8; // lanes 0 & 16
                float4 o0 = { z0, x1, z2, x3 };
                float4 o1 = { z4, x5, z0, x1 };     // k6,k7 duplicate k0,k1 (q = k%6)
                *(float4*)&out[ee * 8 + 0] = o0;
                *(float4*)&out[ee * 8 + 4] = o1;
            }
        }
    }
}

// ---------------------------------------------------------------------------
extern "C" void kernel_launch(void* const* d_in, const int* in_sizes, int n_in,
                              void* d_out, int out_size, void* d_ws, size_t ws_size,
                              hipStream_t stream) {
    const float* x     = (const float*)d_in[0];   // [131072, 24]
    const float* theta = (const float*)d_in[1];   // [3, 6, 3]
    float* out = (float*)d_out;                   // [131072, 8]
    float* G   = (float*)d_ws;                    // 128*128 floats (64 KB)

    build_G_kernel<<<1, 64, 0, stream>>>(theta, G);
    qsim_kernel<<<131072 / 128, 128, 0, stream>>>(x, G, out);
}